// Net_25168508354613
// MI455X (gfx1250) — compile-verified
//
#include <hip/hip_runtime.h>
#include <cstdint>

// ---------------- problem constants ----------------
constexpr int BATCH = 4;
constexpr int NA    = 3;               // anchors per scale
constexpr int NCLS  = 80;
constexpr int CCH   = NA * (5 + NCLS); // 255 channels
constexpr int H1 = 52, H2 = 26, H3 = 13;
constexpr int N1 = H1 * H1 * NA;       // 8112
constexpr int N2 = H2 * H2 * NA;       // 2028
constexpr int N3 = H3 * H3 * NA;       // 507
constexpr int NBOX = N1 + N2 + N3;     // 10647
constexpr int TJ    = 512;             // j-tile size (boxes)
constexpr int TILES = 21;              // ceil(10647/512)
constexpr int NPAD  = TILES * TJ;      // 10752 (padded, conf=0 pads)
constexpr float CONF_TH = 0.5f;
constexpr float IOU_TH  = 0.4f;

__constant__ float ANC[9][2] = {
    {10.f, 13.f}, {16.f, 30.f}, {33.f, 23.f},     // scale1 (52x52)
    {30.f, 61.f}, {62.f, 45.f}, {59.f, 119.f},    // scale2 (26x26)
    {116.f, 90.f}, {156.f, 198.f}, {373.f, 326.f} // scale3 (13x13)
};

__device__ __forceinline__ float sigmoidf(float x) {
    return 1.0f / (1.0f + __expf(-x));
}

// -------------------------------------------------------------------
// Kernel 1: decode all 3 scales.  One thread per (batch, padded box).
// Writes det rows [b0,b1,b2,b3,conf,cls] into d_out, and a packed
// 8-float record {b0,b1,b2,b3,conf,area,cls,0} per box into workspace.
// -------------------------------------------------------------------
__global__ __launch_bounds__(256) void decode_kernel(
    const float* __restrict__ s1, const float* __restrict__ s2,
    const float* __restrict__ s3, float* __restrict__ det,
    float* __restrict__ boxes) {
    int gid = blockIdx.x * 256 + threadIdx.x;
    if (gid >= BATCH * NPAD) return;
    int b = gid / NPAD;
    int n = gid % NPAD;
    float* box = boxes + (size_t)gid * 8;

    if (n >= NBOX) { // zero pad: conf=0 -> never valid, never suppresses
        #pragma unroll
        for (int k = 0; k < 8; ++k) box[k] = 0.0f;
        return;
    }

    const float* src;
    int HS, nl, abase;
    if (n < N1)            { src = s1; HS = H1; nl = n;            abase = 0; }
    else if (n < N1 + N2)  { src = s2; HS = H2; nl = n - N1;       abase = 3; }
    else                   { src = s3; HS = H3; nl = n - N1 - N2;  abase = 6; }
    const float stride = 416.0f / (float)HS; // 8 / 16 / 32 exact

    int a  = nl % NA;
    int t  = nl / NA;
    int hh = t % HS;   // cy
    int ww = t / HS;   // cx
    int cs = HS * HS;  // channel stride
    size_t base = (((size_t)b * CCH + a * (5 + NCLS)) * HS + hh) * HS + ww;

    float f0 = src[base];
    float f1 = src[base + (size_t)cs];
    float f2 = src[base + 2 * (size_t)cs];
    float f3 = src[base + 3 * (size_t)cs];
    float f4 = src[base + 4 * (size_t)cs];

    float x = (sigmoidf(f0) + (float)ww) * stride;
    float y = (sigmoidf(f1) + (float)hh) * stride;
    float wd = __expf(f2) * ANC[abase + a][0] * stride;
    float ht = __expf(f3) * ANC[abase + a][1] * stride;
    float conf = sigmoidf(f4);

    // argmax over class logits (sigmoid is monotonic; '>' keeps first max)
    float best = src[base + 5 * (size_t)cs];
    int cls = 0;
    for (int k = 1; k < NCLS; ++k) {
        float v = src[base + (5 + k) * (size_t)cs];
        if (v > best) { best = v; cls = k; }
    }

    float b0 = x - wd * 0.5f;
    float b1 = x + wd * 0.5f;
    float b2 = y - ht * 0.5f;
    float b3 = y + ht * 0.5f;
    float area = fmaxf(b2 - b0 + 1.0f, 0.0f) * fmaxf(b3 - b1 + 1.0f, 0.0f);

    float* d = det + ((size_t)b * NBOX + n) * 6;
    d[0] = b0; d[1] = b1; d[2] = b2; d[3] = b3; d[4] = conf; d[5] = (float)cls;

    box[0] = b0; box[1] = b1; box[2] = b2; box[3] = b3;
    box[4] = conf; box[5] = area; box[6] = (float)cls; box[7] = 0.0f;
}

// -------------------------------------------------------------------
// Async global->LDS staging of one 512-box tile (16 KB).
// 128 threads x 8 issues; each issue moves 32 lanes x 16B per wave.
// GVS addressing: mem = SADDR(sgpr pair) + VADDR(u32) ; LDS dst in VDST.
// -------------------------------------------------------------------
__device__ __forceinline__ void issue_tile(const float* gbase, uint32_t gbyte,
                                           uint32_t ldsbase, int tid) {
    #pragma unroll
    for (int q = 0; q < 8; ++q) {
        uint32_t off = (uint32_t)((q * 128 + tid) * 16);
        asm volatile("global_load_async_to_lds_b128 %0, %1, %2"
                     :
                     : "v"(ldsbase + off), "v"(gbyte + off), "s"(gbase)
                     : "memory");
    }
}

// -------------------------------------------------------------------
// Kernel 2: pairwise NMS.  grid = (ceil(N/128), B), 128 threads/block.
// Thread owns box i; sweeps all j via double-buffered async LDS tiles.
// keep[i] = valid[i] && any_j( iou>0.4 && same class && conf_i<conf_j
//                              && valid[j] )     (reference semantics)
// Block exits early once every lane has resolved its existential test.
// -------------------------------------------------------------------
__global__ __launch_bounds__(128) void nms_kernel(
    const float* __restrict__ boxes, float* __restrict__ keep_out) {
    __shared__ __align__(16) float sh[2][TJ * 8]; // 2 x 16 KB

    const int tid = threadIdx.x;
    const int b   = blockIdx.y;
    const int i   = blockIdx.x * 128 + tid;
    const float* bb = boxes + (size_t)b * NPAD * 8;

    const uint32_t lds0 = (uint32_t)(uintptr_t)&sh[0][0];
    const uint32_t lds1 = (uint32_t)(uintptr_t)&sh[1][0];

    float mb0 = 0.f, mb1 = 0.f, mb2 = 0.f, mb3 = 0.f;
    float mconf = -1.f, marea = 0.f, mcls = -1.f;
    const bool activeI = (i < NBOX);
    if (activeI) {
        const float4* p = (const float4*)(bb + (size_t)i * 8);
        float4 q0 = p[0], q1 = p[1];
        mb0 = q0.x; mb1 = q0.y; mb2 = q0.z; mb3 = q0.w;
        mconf = q1.x; marea = q1.y; mcls = q1.z;
    }
    const bool valid = activeI && (mconf > CONF_TH);
    bool found = false;

    issue_tile(bb, 0u, lds0, tid); // prefetch tile 0

    for (int t = 0; t < TILES; ++t) {
        const float4* shv = (t & 1) ? (const float4*)&sh[1][0]
                                    : (const float4*)&sh[0][0];
        if (t + 1 < TILES) {
            issue_tile(bb, (uint32_t)((t + 1) * TJ * 32),
                       (t & 1) ? lds0 : lds1, tid);
            // the 8 just-issued loads stay in flight; wait for prior tile only
            asm volatile("s_wait_asynccnt 8" ::: "memory");
        } else {
            asm volatile("s_wait_asynccnt 0" ::: "memory");
        }
        __syncthreads();

        if (valid && !found) {
            #pragma unroll 4
            for (int j = 0; j < TJ; ++j) {           // broadcast LDS reads
                float4 q0 = shv[j * 2 + 0];          // b0 b1 b2 b3
                float4 q1 = shv[j * 2 + 1];          // conf area cls pad
                float ixmin = fmaxf(mb0, q0.x);
                float iymin = fmaxf(mb1, q0.y);
                float ixmax = fminf(mb2, q0.z);
                float iymax = fminf(mb3, q0.w);
                float iw = fmaxf(ixmax - ixmin + 1.0f, 0.0f);
                float ih = fmaxf(iymax - iymin + 1.0f, 0.0f);
                float inter = iw * ih;
                float denom = marea + q1.y - inter;  // >= 0 (inter<=min area)
                bool supp = (inter > IOU_TH * denom) & (mcls == q1.z) &
                            (mconf < q1.x) & (q1.x > CONF_TH);
                found |= supp;
            }
        }

        // Barrier (protects buffer reuse) + block-wide vote: exit as soon as
        // every lane has either found a suppressor or is invalid.  Any async
        // loads still in flight are drained by S_ENDPGM's implicit wait-idle.
        int done = (found || !valid) ? 1 : 0;
        if (__syncthreads_and(done)) break;
    }

    if (activeI) keep_out[(size_t)b * NBOX + i] = (valid && found) ? 1.0f : 0.0f;
}

// -------------------------------------------------------------------
extern "C" void kernel_launch(void* const* d_in, const int* in_sizes, int n_in,
                              void* d_out, int out_size, void* d_ws,
                              size_t ws_size, hipStream_t stream) {
    const float* s1 = (const float*)d_in[0]; // (4,255,52,52)
    const float* s2 = (const float*)d_in[1]; // (4,255,26,26)
    const float* s3 = (const float*)d_in[2]; // (4,255,13,13)

    float* det  = (float*)d_out;                         // (B,N,6)
    float* keep = det + (size_t)BATCH * NBOX * 6;        // (B,N) as 0/1
    float* boxes = (float*)d_ws;                         // (B,NPAD,8) = 1.31 MB

    {
        int total = BATCH * NPAD;
        dim3 grid((total + 255) / 256), block(256);
        decode_kernel<<<grid, block, 0, stream>>>(s1, s2, s3, det, boxes);
    }
    {
        dim3 grid((NBOX + 127) / 128, BATCH), block(128);
        nms_kernel<<<grid, block, 0, stream>>>(boxes, keep);
    }
}